// unit_gcn_45981919871743
// MI455X (gfx1250) — compile-verified
//
#include <hip/hip_runtime.h>
#include <hip/hip_bf16.h>

typedef __attribute__((ext_vector_type(2))) float v2f;
typedef __attribute__((ext_vector_type(8))) float v8f;

// Problem constants
#define NB   128
#define CIN  64
#define COUT 64
#define TDIM 256
#define VJ   25
#define SSUB 3
#define RED  4
#define TT   4            // t-tile per block
#define NTILES (TDIM/TT)  // 64
#define XSTR 105          // xs row stride (odd -> conflict free)
#define WFSTR 196         // wf row stride (196 mod 64 = 4)
#define XAW  26           // xa row width (25 + 1 pad)

// Workspace layout (floats)
#define WS_AF    0          // 3*25*25 = 1875 (pad to 1920)
#define WS_WF    1920       // 3*64*64 = 12288
#define WS_BIAS  14208      // 64
#define WS_P     14272      // 128*64*25 = 204800
#define WS_G     219072     // 128*64*25 = 204800

// ---------------------------------------------------------------------------
// Kernel 1: fold BN into weights, build Af = A + PA/||PA||, effective bias.
// ---------------------------------------------------------------------------
__global__ void prep_kernel(const float* __restrict__ A, const float* __restrict__ PA,
                            const float* __restrict__ wd, const float* __restrict__ bd,
                            const float* __restrict__ gam, const float* __restrict__ bet,
                            const float* __restrict__ mean, const float* __restrict__ var,
                            float* __restrict__ Afg, float* __restrict__ wfoldg,
                            float* __restrict__ biasg) {
    __shared__ float normS[SSUB * VJ];
    __shared__ float scaleS[COUT];
    __shared__ float shiftS[COUT];
    int tid = threadIdx.x;
    if (tid < SSUB * VJ) {                 // per (s, w) column norm over v
        int s = tid / VJ, w = tid % VJ;
        float acc = 0.f;
        for (int v = 0; v < VJ; ++v) {
            float pv = PA[(s * VJ + v) * VJ + w];
            acc += pv * pv;
        }
        normS[tid] = sqrtf(acc) + 1e-4f;
    }
    if (tid >= 128 && tid < 128 + COUT) {  // BN scale/shift
        int o = tid - 128;
        float sc = gam[o] * rsqrtf(var[o] + 1e-5f);
        scaleS[o] = sc;
        shiftS[o] = bet[o] - mean[o] * sc;
    }
    __syncthreads();
    for (int f = tid; f < SSUB * VJ * VJ; f += 256) {
        int s = f / (VJ * VJ), r = f % (VJ * VJ), w = r % VJ;
        Afg[f] = A[f] + PA[f] / normS[s * VJ + w];
    }
    for (int f = tid; f < SSUB * COUT * CIN; f += 256) {
        int o = (f >> 6) & 63;             // f = (s*64 + o)*64 + c
        wfoldg[f] = wd[f] * scaleS[o];
    }
    if (tid < COUT) {
        float b = bd[tid] + bd[COUT + tid] + bd[2 * COUT + tid];
        biasg[tid] = b * scaleS[tid] + shiftS[tid];
    }
}

// ---------------------------------------------------------------------------
// Kernel 2: main fused GCN using V_WMMA_F32_16X16X4_F32.
//   Stage A: xA[s,(c,t),w] = sum_v x[c,t,v] * Af[s,v,w]   (K = 25 -> 28)
//   Stage B: y[o,(t,w)]    = sum_{s,c} wfold[o,(s,c)] * xA (K = 192)
//   Epilogue: + bias_eff + residual x, ReLU, store y.
// ---------------------------------------------------------------------------
__global__ void __launch_bounds__(256)
gcn_wmma_kernel(const float* __restrict__ x, const float* __restrict__ Afg,
                const float* __restrict__ wfoldg, const float* __restrict__ biasg,
                float* __restrict__ out) {
    __shared__ float xs[CIN * XSTR];            // x tile  [c][t*25+v]        6720 f
    __shared__ float afl[SSUB * 28 * 32];       // Af padded [s][v(28)][w(32)] 2688 f
    __shared__ float wf[COUT * WFSTR];          // wfold [o][s*64+c]          12544 f
    __shared__ float xa[SSUB * 256 * XAW];      // xA [s][m=(c,t)][w]         19968 f
    __shared__ float biasS[COUT];

    const int tid  = threadIdx.x;
    const int lane = tid & 31;
    const int wave = tid >> 5;
    const int n  = blockIdx.x >> 6;             // batch sample
    const int t0 = (blockIdx.x & 63) * TT;      // t-tile origin

    // ---- zero pads ----
    for (int f = tid; f < CIN * XSTR; f += 256) xs[f] = 0.f;
    for (int f = tid; f < SSUB * 28 * 32; f += 256) afl[f] = 0.f;
    __syncthreads();

    // ---- fill LDS ----
    // x tile: per c, floats [t0*25 .. t0*25+99] are contiguous in global memory
    for (int f = tid; f < CIN * (TT * VJ); f += 256) {
        int c = f / (TT * VJ), off = f % (TT * VJ);
        xs[c * XSTR + off] = x[((size_t)(n * CIN + c) * TDIM + t0) * VJ + off];
    }
    for (int f = tid; f < SSUB * VJ * VJ; f += 256) {
        int s = f / (VJ * VJ), r = f % (VJ * VJ), v = r / VJ, w = r % VJ;
        afl[(s * 28 + v) * 32 + w] = Afg[f];
    }
    for (int f = tid; f < SSUB * COUT * CIN; f += 256) {
        int s = f / (COUT * CIN), r = f % (COUT * CIN), o = r / CIN, c = r % CIN;
        wf[o * WFSTR + s * 64 + c] = wfoldg[f];
    }
    if (tid < COUT) biasS[tid] = biasg[tid];
    __syncthreads();

    // ---- Stage A: 96 tile jobs = s(3) x mtile(16) x ntile(2), K = 28 ----
    const int h   = lane >> 4;      // half selector (K offset)
    const int l15 = lane & 15;
    for (int j = wave; j < 96; j += 8) {
        int s = j >> 5, rem = j & 31, mtile = rem >> 1, ntile = rem & 1;
        int mbase = mtile * 16;
        int m = mbase + l15;
        int c = m >> 2, t = m & 3;
        // h-dependent bases hoisted so unrolled offsets are immediates
        const float* Arow = &xs[c * XSTR + t * VJ + 2 * h];       // A[m][k0+2h]
        int nb = ntile * 16 + l15;                                // B column (w)
        const float* Brow = &afl[(s * 28 + 2 * h) * 32 + nb];     // B[k0+2h][nb]
        v8f acc = {0.f, 0.f, 0.f, 0.f, 0.f, 0.f, 0.f, 0.f};
#pragma unroll
        for (int k0 = 0; k0 < 28; k0 += 4) {
            v2f a, b;
            a.x = Arow[k0];
            a.y = Arow[k0 + 1];
            b.x = Brow[k0 * 32];
            b.y = Brow[k0 * 32 + 32];
            acc = __builtin_amdgcn_wmma_f32_16x16x4_f32(
                false, a, false, b, (short)0, acc, false, false);
        }
        int wcol = ntile * 16 + l15;                  // D column of this lane
        if (wcol < VJ) {
#pragma unroll
            for (int r = 0; r < 8; ++r) {
                int mr = mbase + r + 8 * h;           // D row held in VGPR r
                xa[(s * 256 + mr) * XAW + wcol] = acc[r];
            }
        }
    }
    __syncthreads();

    // ---- Stage B: 28 tile jobs = mtile(4,o) x ntile(7,(t,w)), K = 192 ----
    for (int j = wave; j < 28; j += 8) {
        int mtile = j / 7, ntile = j % 7;
        int o = mtile * 16 + l15;
        const float* Wrow = &wf[o * WFSTR + 2 * h];   // A[o][k0+2h]
        int col = ntile * 16 + l15;
        int jc = col < 100 ? col : 99;                // clamp pad lanes to safe addr
        int tl = jc / VJ, wl = jc % VJ;
        // B element (k, col): addr = ((k>>6)*256 + (k&63)*4 + tl)*26 + wl.
        // With k = k0 + 2h (k0 % 4 == 0, 2h <= 2): k>>6 == k0>>6 and
        // k&63 == (k0&63) + 2h, so addr = base(h,tl,wl) + const(k0).
        const float* xab = &xa[tl * XAW + wl + (2 * h) * 4 * XAW];
        v8f acc = {0.f, 0.f, 0.f, 0.f, 0.f, 0.f, 0.f, 0.f};
#pragma unroll
        for (int k0 = 0; k0 < 192; k0 += 4) {
            int c0 = (k0 >> 6) * (256 * XAW) + (k0 & 63) * (4 * XAW); // folds to const
            v2f a, b;
            a.x = Wrow[k0];
            a.y = Wrow[k0 + 1];
            b.x = xab[c0];            // k = k0 + 2h
            b.y = xab[c0 + 4 * XAW];  // k = k0 + 2h + 1
            acc = __builtin_amdgcn_wmma_f32_16x16x4_f32(
                false, a, false, b, (short)0, acc, false, false);
        }
        // Epilogue: bias + residual + ReLU -> global y
        if (col < 100) {
            int tt = col / VJ, ww = col % VJ;
#pragma unroll
            for (int r = 0; r < 8; ++r) {
                int oo = mtile * 16 + r + 8 * h;
                float val = acc[r] + biasS[oo] + xs[oo * XSTR + tt * VJ + ww];
                val = fmaxf(val, 0.f);
                out[((size_t)(n * COUT + oo) * TDIM + (t0 + tt)) * VJ + ww] = val;
            }
        }
    }
}

// ---------------------------------------------------------------------------
// Kernel 3: deterministic temporal mean pool  p[n,o,v] = mean_t y[n,o,t,v]
// ---------------------------------------------------------------------------
__global__ void pool_kernel(const float* __restrict__ y, float* __restrict__ p) {
    __shared__ float part[8][VJ];
    int no = blockIdx.x;                       // n*64 + o
    int tid = threadIdx.x;
    if (tid < 200) {
        int g = tid / VJ, v = tid % VJ;        // 8 t-groups x 25 joints
        const float* base = y + (size_t)no * (TDIM * VJ);
        float s = 0.f;
        for (int t = g * 32; t < g * 32 + 32; ++t) s += base[t * VJ + v];
        part[g][v] = s;
    }
    __syncthreads();
    if (tid < VJ) {
        float s = 0.f;
#pragma unroll
        for (int g = 0; g < 8; ++g) s += part[g][tid];
        p[no * VJ + tid] = s * (1.0f / (float)TDIM);
    }
}

// ---------------------------------------------------------------------------
// Kernel 4: SCE — conv1d(V, ker=25, pad=12) -> relu -> 1x1 -> 1 + sigmoid
// ---------------------------------------------------------------------------
__global__ void sce_kernel(const float* __restrict__ p, const float* __restrict__ sws,
                           const float* __restrict__ sbs, const float* __restrict__ swe,
                           const float* __restrict__ sbe, float* __restrict__ gp1) {
    __shared__ float pL[COUT * VJ];
    __shared__ float sL[RED * VJ];
    int n = blockIdx.x, tid = threadIdx.x;
    for (int f = tid; f < COUT * VJ; f += 128) pL[f] = p[n * COUT * VJ + f];
    __syncthreads();
    if (tid < RED * VJ) {
        int r = tid / VJ, vo = tid % VJ;
        float acc = sbs[r];
        for (int o = 0; o < COUT; ++o) {
            const float* wrow = sws + (r * COUT + o) * VJ;
            const float* prow = pL + o * VJ;
            for (int k = 0; k < VJ; ++k) {
                int vi = vo + k - 12;
                if (vi >= 0 && vi < VJ) acc += prow[vi] * wrow[k];
            }
        }
        sL[tid] = fmaxf(acc, 0.f);
    }
    __syncthreads();
    for (int f = tid; f < COUT * VJ; f += 128) {
        int o = f / VJ, v = f % VJ;
        float z = sbe[o];
#pragma unroll
        for (int r = 0; r < RED; ++r) z += sL[r * VJ + v] * swe[o * RED + r];
        gp1[n * COUT * VJ + f] = 1.0f + 1.0f / (1.0f + __expf(-z));
    }
}

// ---------------------------------------------------------------------------
// Kernel 5: in-place gating  out[n,o,t,v] *= gp1[n,o,v]
// ---------------------------------------------------------------------------
__global__ void gate_kernel(float* __restrict__ out, const float* __restrict__ gp1,
                            long long total) {
    long long i = (long long)blockIdx.x * blockDim.x + threadIdx.x;
    if (i < total) {
        int v = (int)(i % VJ);
        long long no = i / (long long)(TDIM * VJ);
        out[i] *= gp1[no * VJ + v];
    }
}

extern "C" void kernel_launch(void* const* d_in, const int* in_sizes, int n_in,
                              void* d_out, int out_size, void* d_ws, size_t ws_size,
                              hipStream_t stream) {
    (void)in_sizes; (void)n_in; (void)out_size; (void)ws_size;
    const float* x    = (const float*)d_in[0];
    const float* A    = (const float*)d_in[1];
    const float* PA   = (const float*)d_in[2];
    const float* wd   = (const float*)d_in[3];
    const float* bd   = (const float*)d_in[4];
    const float* gam  = (const float*)d_in[5];
    const float* bet  = (const float*)d_in[6];
    const float* mean = (const float*)d_in[7];
    const float* var  = (const float*)d_in[8];
    const float* sws  = (const float*)d_in[9];
    const float* sbs  = (const float*)d_in[10];
    const float* swe  = (const float*)d_in[11];
    const float* sbe  = (const float*)d_in[12];
    float* out = (float*)d_out;
    float* ws  = (float*)d_ws;

    float* Afg    = ws + WS_AF;
    float* wfoldg = ws + WS_WF;
    float* biasg  = ws + WS_BIAS;
    float* p      = ws + WS_P;
    float* gp1    = ws + WS_G;

    prep_kernel<<<1, 256, 0, stream>>>(A, PA, wd, bd, gam, bet, mean, var,
                                       Afg, wfoldg, biasg);
    gcn_wmma_kernel<<<NB * NTILES, 256, 0, stream>>>(x, Afg, wfoldg, biasg, out);
    pool_kernel<<<NB * COUT, 256, 0, stream>>>(out, p);
    sce_kernel<<<NB, 128, 0, stream>>>(p, sws, sbs, swe, sbe, gp1);

    long long total = (long long)NB * COUT * TDIM * VJ;   // 52,428,800
    int gblocks = (int)((total + 255) / 256);
    gate_kernel<<<gblocks, 256, 0, stream>>>(out, gp1, total);
}